// LambdaRankLoss_10282151707682
// MI455X (gfx1250) — compile-verified
//
#include <hip/hip_runtime.h>
#include <hip/hip_bf16.h>

typedef __attribute__((ext_vector_type(2))) float v2f;
typedef __attribute__((ext_vector_type(8))) float v8f;

#define NSEG   16     // j-dimension split for occupancy
#define KDCG   512

// ---------------------------------------------------------------------------
// Pair term (without the global Ninv scale, applied at the end):
//   lam_ij = (0.5*(1-Sij) - 1/(1+exp(p_i-p_j))) * | (2^ti - 2^tj) * (d_i - d_j) |
// sign(t_i-t_j) == sign(2^ti - 2^tj); when equal, the |.| factor is 0 anyway.
// ---------------------------------------------------------------------------
__device__ __forceinline__ float lam_pair(float pi, float gi, float di, float4 q) {
    float dp = pi - q.x;                       // p_i - p_j
    float gd = gi - q.y;                       // 2^ti - 2^tj
    float dd = di - q.z;                       // decay_i - decay_j
    float e  = __expf(dp);
    float s  = __builtin_amdgcn_rcpf(1.0f + e);     // 1/(1+exp(dp))
    float sel = (gd < 0.0f) ? 1.0f : 0.0f;          // 0.5*(1-sign(gd)), exact when gd!=0
    return (sel - s) * fabsf(gd * dd);
}

// ---------------------------------------------------------------------------
// Kernel 1: per-chunk (32 elems) stable-rank-among-equals + per-value counts.
// target values are exactly {0,1,2,3,4} as floats.
// ---------------------------------------------------------------------------
__global__ void k_hist(const float* __restrict__ target,
                       int* __restrict__ localeq,
                       int* __restrict__ chunkcnt, int n) {
    __shared__ float ts[256];
    int gid = blockIdx.x * 256 + threadIdx.x;
    float t = (gid < n) ? target[gid] : -1.0f;
    ts[threadIdx.x] = t;
    __syncthreads();
    if (gid >= n) return;
    int lane = threadIdx.x & 31;
    int base = threadIdx.x & ~31;
    int cnt = 0;
    for (int k = 0; k < lane; ++k) cnt += (ts[base + k] == t) ? 1 : 0;
    localeq[gid] = cnt;
    if (lane < 5) {                       // lanes 0..4 count value==lane in chunk
        float fv = (float)lane;
        int c = 0;
        for (int k = 0; k < 32; ++k) c += (ts[base + k] == fv) ? 1 : 0;
        int chunk = gid >> 5;
        chunkcnt[chunk * 5 + lane] = c;
    }
}

// ---------------------------------------------------------------------------
// Kernel 2 (single block, 512 thr): chunk-prefix scans, value offsets, maxDCG.
// ---------------------------------------------------------------------------
__global__ void k_scan_dcg(const int* __restrict__ chunkcnt,
                           int* __restrict__ chunkoff,
                           int* __restrict__ valueoff,
                           float* __restrict__ ninv_out,
                           int nchunks, int n) {
    __shared__ int   totals[5];
    __shared__ float red[512];
    int tid = threadIdx.x;
    if (tid < 5) {                        // serial exclusive scan per value (256 iters)
        int run = 0;
        for (int c = 0; c < nchunks; ++c) {
            chunkoff[c * 5 + tid] = run;
            run += chunkcnt[c * 5 + tid];
        }
        totals[tid] = run;
    }
    __syncthreads();
    if (tid == 0) {
        int off = 0;
        for (int v = 0; v < 5; ++v) { valueoff[v] = off; off += totals[v]; }
    }
    // max DCG over top-k of descending-sorted targets
    int k = (n < KDCG) ? n : KDCG;
    int c4 = totals[4];
    int c3 = c4 + totals[3];
    int c2 = c3 + totals[2];
    int c1 = c2 + totals[1];
    float term = 0.0f;
    if (tid < k) {
        float v;
        if      (tid < c4) v = 4.0f;
        else if (tid < c3) v = 3.0f;
        else if (tid < c2) v = 2.0f;
        else if (tid < c1) v = 1.0f;
        else               v = 0.0f;
        float gain = exp2f(v) - 1.0f;
        float disc = __log2f((float)(tid + 2));
        term = gain / disc;
    }
    red[tid] = term;
    __syncthreads();
    for (int s = 256; s > 0; s >>= 1) {
        if (tid < s) red[tid] += red[tid + s];
        __syncthreads();
    }
    if (tid == 0) *ninv_out = 1.0f / red[0];
}

// ---------------------------------------------------------------------------
// Kernel 3: pack per-element {p, 2^t, decay} as float4 for b128 streaming.
// rank0 = (#targets < t) + (#equal targets before i); decay = 1/log2(rank0+2)
// ---------------------------------------------------------------------------
__global__ void k_pack(const float* __restrict__ pred,
                       const float* __restrict__ target,
                       const int* __restrict__ localeq,
                       const int* __restrict__ chunkoff,
                       const int* __restrict__ valueoff,
                       float4* __restrict__ jd, int n) {
    int gid = blockIdx.x * 256 + threadIdx.x;
    if (gid >= n) return;
    float t = target[gid];
    int v = (int)t;
    int chunk = gid >> 5;
    int rank0 = valueoff[v] + chunkoff[chunk * 5 + v] + localeq[gid];
    float4 q;
    q.x = pred[gid];
    q.y = exp2f(t);
    q.z = 1.0f / __log2f((float)(rank0 + 2));
    q.w = 0.0f;
    jd[gid] = q;
}

// ---------------------------------------------------------------------------
// Kernel 4: O(N^2) pair kernel. One wave owns TWO 16-row tiles (32 rows) so
// each pair of j-side b128 loads feeds 4 lam evals/lane and 2 WMMAs.
// Each lane fills the 16x4 f32 A-tile slots (lane<16 -> K{0,1}, lane>=16 ->
// K{2,3}); v_wmma_f32_16x16x4_f32 vs an all-ones B accumulates exact f32 row
// sums in C on the matrix pipe, co-executing with the VALU sigmoid work.
// C layout: lane 0 holds (M=0..7, N=0) in c[0..7]; lane 16 holds M=8..15.
// ---------------------------------------------------------------------------
__global__ void __launch_bounds__(256)
k_pairs(const float4* __restrict__ jd,
        float* __restrict__ partials, int n, int seglen) {
    int lane    = threadIdx.x & 31;
    int wave    = threadIdx.x >> 5;
    int rowBase = (blockIdx.x * 8 + wave) * 32;   // two 16-row tiles
    int m       = lane & 15;

    float4 r0 = jd[rowBase + m];                  // tile 0 row data
    float4 r1 = jd[rowBase + 16 + m];             // tile 1 row data
    float p0 = r0.x, g0 = r0.y, d0 = r0.z;
    float p1 = r1.x, g1 = r1.y, d1 = r1.z;

    v8f c0 = {0.f, 0.f, 0.f, 0.f, 0.f, 0.f, 0.f, 0.f};
    v8f c1 = {0.f, 0.f, 0.f, 0.f, 0.f, 0.f, 0.f, 0.f};
    v2f b; b.x = 1.0f; b.y = 1.0f;                // ones B (4x16) -> row reduction

    int koff     = (lane >> 4) << 1;              // 0 for lanes 0-15, 2 for 16-31
    int segStart = blockIdx.y * seglen;
    int segEnd   = segStart + seglen;

    for (int jj = segStart; jj < segEnd; jj += 4) {
        int j0 = jj + koff;
        float4 q0 = jd[j0];
        float4 q1 = jd[j0 + 1];
        v2f a0, a1;
        a0.x = lam_pair(p0, g0, d0, q0);          // tile0: A[m, koff]
        a0.y = lam_pair(p0, g0, d0, q1);          // tile0: A[m, koff+1]
        a1.x = lam_pair(p1, g1, d1, q0);          // tile1: A[m, koff]
        a1.y = lam_pair(p1, g1, d1, q1);          // tile1: A[m, koff+1]
        // 8-arg f32 WMMA: (neg_a, A, neg_b, B, c_mod, C, reuse_a, reuse_b)
        c0 = __builtin_amdgcn_wmma_f32_16x16x4_f32(false, a0, false, b,
                                                   (short)0, c0, false, false);
        c1 = __builtin_amdgcn_wmma_f32_16x16x4_f32(false, a1, false, b,
                                                   (short)0, c1, false, false);
    }

    if ((lane & 15) == 0) {                       // lanes 0 and 16: N=0 column
        int half = (lane >> 4) << 3;              // 0 or 8
        float* outp = partials + (size_t)blockIdx.y * n + rowBase + half;
#pragma unroll
        for (int r = 0; r < 8; ++r) outp[r]      = c0[r];
#pragma unroll
        for (int r = 0; r < 8; ++r) outp[16 + r] = c1[r];
    }
}

// ---------------------------------------------------------------------------
// Kernel 5: deterministic fixed-order segment reduction + Ninv scale.
// ---------------------------------------------------------------------------
__global__ void k_reduce(const float* __restrict__ partials,
                         const float* __restrict__ ninv_p,
                         float* __restrict__ out, int n, int nseg) {
    int gid = blockIdx.x * 256 + threadIdx.x;
    if (gid >= n) return;
    float s = 0.0f;
    for (int q = 0; q < nseg; ++q) s += partials[(size_t)q * n + gid];
    out[gid] = (*ninv_p) * s;
}

// ---------------------------------------------------------------------------
extern "C" void kernel_launch(void* const* d_in, const int* in_sizes, int n_in,
                              void* d_out, int out_size, void* d_ws, size_t ws_size,
                              hipStream_t stream) {
    const float* pred = (const float*)d_in[0];
    const float* targ = (const float*)d_in[1];
    int n = in_sizes[0];                       // 8192
    int nchunks = n >> 5;                      // 256

    // workspace carve-up (16B aligned pieces)
    char*   w        = (char*)d_ws;
    float4* jd       = (float4*)w;                                    // n*16 B
    float*  partials = (float*)(w + (size_t)n * 16);                  // NSEG*n*4 B
    int*    localeq  = (int*)(w + (size_t)n * 16 + (size_t)NSEG * n * 4);
    int*    chunkcnt = localeq + n;                                   // nchunks*5
    int*    chunkoff = chunkcnt + nchunks * 5;                        // nchunks*5
    int*    valueoff = chunkoff + nchunks * 5;                        // 5
    float*  ninv_p   = (float*)(valueoff + 8);                        // 1

    int blocks256 = (n + 255) / 256;

    k_hist<<<blocks256, 256, 0, stream>>>(targ, localeq, chunkcnt, n);
    k_scan_dcg<<<1, 512, 0, stream>>>(chunkcnt, chunkoff, valueoff, ninv_p, nchunks, n);
    k_pack<<<blocks256, 256, 0, stream>>>(pred, targ, localeq, chunkoff, valueoff, jd, n);

    int seglen = n / NSEG;                     // 512
    dim3 grid(n / 256, NSEG);                  // 8 waves/block * 32 rows = 256 rows/block
    k_pairs<<<grid, 256, 0, stream>>>(jd, partials, n, seglen);

    k_reduce<<<blocks256, 256, 0, stream>>>(partials, ninv_p, (float*)d_out, n, NSEG);
}